// LMMD_loss_56495999812246
// MI455X (gfx1250) — compile-verified
//
#include <hip/hip_runtime.h>

typedef __attribute__((ext_vector_type(2))) float v2f;
typedef __attribute__((ext_vector_type(8))) float v8f;
typedef __attribute__((ext_vector_type(4))) unsigned int u32x4;
typedef __attribute__((ext_vector_type(8))) int i32x8;
typedef __attribute__((ext_vector_type(4))) int i32x4;

#define BATCH 4096
#define DIM 256
#define N_TOT 8192
#define CLS 31

#define BTILE 128            // block tile (M and N)
#define KCHUNK 32            // K chunk staged per TDM transfer
#define LDS_STRIDE 34        // 32 data floats + 2 pad floats (TDM pad feature)

// float-indexed workspace layout
#define WS_SUMSQ   0
#define WS_COLSUM  1            // 256 floats
#define WS_C16     257          // log2(e)/(16*bandwidth)
#define WS_LOSS    258
#define WS_NANFLAG 259          // int
#define WS_WSS     260          // 31
#define WS_WTT     291          // 31
#define WS_WST     322          // 31
#define WS_SQ      384          // 8192 floats
#define WS_FLOATS  (WS_SQ + N_TOT)

// ---------------- per-class weights (one block) ----------------
__global__ void k_weights(const int* __restrict__ sl, const int* __restrict__ tl,
                          float* __restrict__ ws) {
    __shared__ int scnt[CLS], tcnt[CLS];
    int t = threadIdx.x;
    if (t < CLS) { scnt[t] = 0; tcnt[t] = 0; }
    __syncthreads();
    for (int i = t; i < BATCH; i += blockDim.x) {
        atomicAdd(&scnt[sl[i]], 1);
        atomicAdd(&tcnt[tl[i]], 1);
    }
    __syncthreads();
    if (t == 0) {
        float length = 0.f;
        for (int c = 0; c < CLS; ++c)
            if (scnt[c] > 0 && tcnt[c] > 0) length += 1.f;
        float scale = (length > 0.f) ? 1.f / fmaxf(length, 1.f) : 0.f;
        for (int c = 0; c < CLS; ++c) {
            bool p = (scnt[c] > 0) && (tcnt[c] > 0);
            float sc = (float)scnt[c], tc = (float)tcnt[c];
            ws[WS_WSS + c] = p ? scale / (sc * sc) : 0.f;
            ws[WS_WTT + c] = p ? scale / (tc * tc) : 0.f;
            ws[WS_WST + c] = p ? scale / (sc * tc) : 0.f;
        }
    }
}

// ---------------- row stats: sq[i], sum(sq), column sums ----------------
__global__ void k_rowstats(const float* __restrict__ src, const float* __restrict__ tgt,
                           float* __restrict__ ws) {
    __shared__ float cs[DIM];
    __shared__ float red[256];
    int t = threadIdx.x;
    cs[t] = 0.f;
    __syncthreads();

    int row = blockIdx.x * blockDim.x + t;
    const float* p = (row < BATCH) ? (src + (size_t)row * DIM)
                                   : (tgt + (size_t)(row - BATCH) * DIM);
    float sq = 0.f;
    for (int d0 = 0; d0 < DIM; ++d0) {
        int d = (d0 + t) & (DIM - 1);          // stagger to dodge LDS bank conflicts
        float v = p[d];
        sq += v * v;
        atomicAdd(&cs[d], v);
    }
    ws[WS_SQ + row] = sq;

    red[t] = sq;
    __syncthreads();
    for (int s = 128; s > 0; s >>= 1) {
        if (t < s) red[t] += red[t + s];
        __syncthreads();
    }
    if (t == 0) atomicAdd(&ws[WS_SUMSQ], red[0]);
    atomicAdd(&ws[WS_COLSUM + t], cs[t]);
}

// ---------------- bandwidth finalize ----------------
__global__ void k_finalize(float* __restrict__ ws) {
    if (threadIdx.x == 0 && blockIdx.x == 0) {
        float sumsq = ws[WS_SUMSQ];
        float sumG = 0.f;
        for (int d = 0; d < DIM; ++d) { float c = ws[WS_COLSUM + d]; sumG += c * c; }
        float n = (float)N_TOT;
        float sum_l2 = 2.f * n * sumsq - 2.f * sumG;     // sum of all pairwise sq dists
        float bw = sum_l2 / (n * n - n);
        bw *= 0.25f;                                     // / kernel_mul^(kernel_num//2)
        ws[WS_C16] = 1.4426950408889634f / (16.f * bw);  // log2(e)/(16*bw)
    }
}

// ---------------- TDM: stage a 128x32 fp32 tile into LDS with row pad ----------------
// D# per CDNA5 ISA: group0 {count, lds_addr, global_addr, type=2},
// group1 {data_size=4B, pad(2 dw / 32 dw), tensor 256 x 4096, tile 32 x 128, stride0=256}.
__device__ __forceinline__ void tdm_load_tile(unsigned lds_off, const float* gptr) {
    unsigned long long ga = (unsigned long long)(uintptr_t)gptr;
    u32x4 g0;
    g0[0] = 1u;                                            // count=1, user mode
    g0[1] = lds_off;                                       // lds_addr (bytes)
    g0[2] = (unsigned)(ga & 0xFFFFFFFFu);                  // global_addr[31:0]
    g0[3] = (unsigned)((ga >> 32) & 0x01FFFFFFu)           // global_addr[56:32]
          | (2u << 30);                                    // type=2 ("image")
    i32x8 g1;
    g1[0] = (2 << 16)        // data_size = 4 bytes
          | (1 << 20)        // pad_enable
          | (4 << 22)        // pad_interval: every 32 DWORDs
          | (1 << 25);       // pad_amount: 2 DWORDs -> row stride 34 floats
    g1[1] = (int)(DIM   << 16);     // tensor_dim0 = 256 (low 16 bits at [79:48])
    g1[2] = (int)(BATCH << 16);     // tensor_dim1 = 4096 (low 16 bits at [111:80])
    g1[3] = (int)(KCHUNK << 16);    // tile_dim0 = 32  ([127:112])
    g1[4] = BTILE;                  // tile_dim1 = 128 ([143:128])
    g1[5] = DIM;                    // tensor_dim0_stride = 256 ([207:160] low 32)
    g1[6] = 0;
    g1[7] = 0;
    i32x4 gz4 = {0, 0, 0, 0};       // 2-D tile: groups 2/3 unused
    i32x8 gz8 = {0, 0, 0, 0, 0, 0, 0, 0};
    __builtin_amdgcn_tensor_load_to_lds(g0, g1, gz4, gz4, gz8, 0);
}

// ---------------- main fused GEMM + gaussian-kernel + weighted reduce ----------------
// grid (32,32,3): z selects SS / TT / ST. 256 thr = 8 waves; block tile 128x128;
// wave w owns a 32x64 tile = 2x4 WMMA accumulators (8 wmma per 6 LDS loads).
// A/B K-chunks staged to LDS by the Tensor Data Mover, double buffered.
__global__ void __launch_bounds__(256) k_mmd(const float* __restrict__ src,
                                             const float* __restrict__ tgt,
                                             const int* __restrict__ sl,
                                             const int* __restrict__ tl,
                                             float* __restrict__ ws) {
    __shared__ float sA[2][BTILE * LDS_STRIDE];
    __shared__ float sB[2][BTILE * LDS_STRIDE];

    const int z = blockIdx.z;
    const float* X; const float* Y; const int* LX; const int* LY;
    const float* wtab; float sign;
    const float* sq = ws + WS_SQ;
    const float* sqx; const float* sqy;
    if (z == 0)      { X = src; Y = src; LX = sl; LY = sl; wtab = ws + WS_WSS; sign =  1.f; sqx = sq;         sqy = sq;         }
    else if (z == 1) { X = tgt; Y = tgt; LX = tl; LY = tl; wtab = ws + WS_WTT; sign =  1.f; sqx = sq + BATCH; sqy = sq + BATCH; }
    else             { X = src; Y = tgt; LX = sl; LY = tl; wtab = ws + WS_WST; sign = -2.f; sqx = sq;         sqy = sq + BATCH; }

    const int lane  = threadIdx.x & 31;
    const int wave  = threadIdx.x >> 5;
    const int m0    = blockIdx.y * BTILE;
    const int n0    = blockIdx.x * BTILE;
    const int wm    = (wave >> 1) * 32;     // wave row offset in block tile
    const int wn    = (wave &  1) * 64;     // wave col offset in block tile
    const int rlo   = lane & 15;
    const int rhalf = lane >> 4;

    // prologue: TDM-stage chunk 0 (wave 0 issues; EXEC-independent scalar DMA)
    if (wave == 0) {
        tdm_load_tile((unsigned)(uintptr_t)&sA[0][0], X + (size_t)m0 * DIM);
        tdm_load_tile((unsigned)(uintptr_t)&sB[0][0], Y + (size_t)n0 * DIM);
        __builtin_amdgcn_s_wait_tensorcnt(0);
    }
    __syncthreads();

    v8f c[2][4];
    #pragma unroll
    for (int i = 0; i < 2; ++i)
        #pragma unroll
        for (int j = 0; j < 4; ++j) c[i][j] = (v8f){};

    for (int ch = 0; ch < DIM / KCHUNK; ++ch) {
        const int cur = ch & 1;
        if ((ch + 1) < DIM / KCHUNK && wave == 0) {     // prefetch next chunk
            const int kc = (ch + 1) * KCHUNK;
            tdm_load_tile((unsigned)(uintptr_t)&sA[cur ^ 1][0], X + (size_t)m0 * DIM + kc);
            tdm_load_tile((unsigned)(uintptr_t)&sB[cur ^ 1][0], Y + (size_t)n0 * DIM + kc);
        }
        const float* As = &sA[cur][0];
        const float* Bs = &sB[cur][0];
        #pragma unroll
        for (int k = 0; k < KCHUNK; k += 4) {
            const int kk = k + rhalf * 2;   // fp32 WMMA operand: K pair by lane[4]
            v2f a0 = *(const v2f*)(As + (wm      + rlo) * LDS_STRIDE + kk);
            v2f a1 = *(const v2f*)(As + (wm + 16 + rlo) * LDS_STRIDE + kk);
            v2f b0 = *(const v2f*)(Bs + (wn      + rlo) * LDS_STRIDE + kk);
            v2f b1 = *(const v2f*)(Bs + (wn + 16 + rlo) * LDS_STRIDE + kk);
            v2f b2 = *(const v2f*)(Bs + (wn + 32 + rlo) * LDS_STRIDE + kk);
            v2f b3 = *(const v2f*)(Bs + (wn + 48 + rlo) * LDS_STRIDE + kk);
            c[0][0] = __builtin_amdgcn_wmma_f32_16x16x4_f32(false, a0, false, b0, (short)0, c[0][0], false, false);
            c[0][1] = __builtin_amdgcn_wmma_f32_16x16x4_f32(false, a0, false, b1, (short)0, c[0][1], false, false);
            c[0][2] = __builtin_amdgcn_wmma_f32_16x16x4_f32(false, a0, false, b2, (short)0, c[0][2], false, false);
            c[0][3] = __builtin_amdgcn_wmma_f32_16x16x4_f32(false, a0, false, b3, (short)0, c[0][3], false, false);
            c[1][0] = __builtin_amdgcn_wmma_f32_16x16x4_f32(false, a1, false, b0, (short)0, c[1][0], false, false);
            c[1][1] = __builtin_amdgcn_wmma_f32_16x16x4_f32(false, a1, false, b1, (short)0, c[1][1], false, false);
            c[1][2] = __builtin_amdgcn_wmma_f32_16x16x4_f32(false, a1, false, b2, (short)0, c[1][2], false, false);
            c[1][3] = __builtin_amdgcn_wmma_f32_16x16x4_f32(false, a1, false, b3, (short)0, c[1][3], false, false);
        }
        __syncthreads();                                  // all waves done reading buf[cur]
        if (wave == 0) __builtin_amdgcn_s_wait_tensorcnt(0);   // next chunk landed
        __syncthreads();                                  // safe to read buf[cur^1]
    }

    // epilogue: l2 -> 5-bandwidth gaussian sum (1 exp + 4 squarings) -> weighted sum
    const float C16 = ws[WS_C16];
    float acc = 0.f;
    int nanf = 0;

    int   ncol[4]; float sqn[4]; int lyn[4];
    #pragma unroll
    for (int j = 0; j < 4; ++j) {
        ncol[j] = n0 + wn + j * 16 + rlo;   // C/D layout: N = lane&15
        sqn[j]  = sqy[ncol[j]];
        lyn[j]  = LY[ncol[j]];
    }

    #pragma unroll
    for (int i = 0; i < 2; ++i) {
        const int mbase = m0 + wm + i * 16 + rhalf * 8;   // C/D: M = vgpr + 8*lane[4]
        #pragma unroll
        for (int r = 0; r < 8; ++r) {
            const int m = mbase + r;
            const float sqm = sqx[m];
            const int lm = LX[m];
            const float wl = wtab[lm];
            #pragma unroll
            for (int j = 0; j < 4; ++j) {
                float l2 = sqm + sqn[j] - 2.f * c[i][j][r];
                float u  = __builtin_amdgcn_exp2f(-l2 * C16);  // widest bandwidth term
                float u2 = u * u, u4 = u2 * u2, u8 = u4 * u4, u16 = u8 * u8;
                float ks = u + u2 + u4 + u8 + u16;             // all 5 bandwidths
                nanf |= (ks != ks);
                if (lm == lyn[j]) acc += wl * ks;
            }
        }
    }

    if (nanf) atomicOr(((int*)ws) + WS_NANFLAG, 1);

    __shared__ float red[256];
    red[threadIdx.x] = acc;
    __syncthreads();
    for (int s = 128; s > 0; s >>= 1) {
        if (threadIdx.x < s) red[threadIdx.x] += red[threadIdx.x + s];
        __syncthreads();
    }
    if (threadIdx.x == 0) atomicAdd(&ws[WS_LOSS], sign * red[0]);
}

// ---------------- output ----------------
__global__ void k_out(const float* __restrict__ ws, float* __restrict__ out) {
    if (threadIdx.x == 0) {
        int nan = ((const int*)ws)[WS_NANFLAG];
        out[0] = nan ? 0.f : ws[WS_LOSS];
    }
}

extern "C" void kernel_launch(void* const* d_in, const int* in_sizes, int n_in,
                              void* d_out, int out_size, void* d_ws, size_t ws_size,
                              hipStream_t stream) {
    const float* src = (const float*)d_in[0];
    const float* tgt = (const float*)d_in[1];
    const int*   sl  = (const int*)d_in[2];
    const int*   tl  = (const int*)d_in[3];
    float* ws  = (float*)d_ws;
    float* out = (float*)d_out;

    (void)hipMemsetAsync(d_ws, 0, WS_FLOATS * sizeof(float), stream);
    k_weights <<<1, 256, 0, stream>>>(sl, tl, ws);
    k_rowstats<<<N_TOT / 256, 256, 0, stream>>>(src, tgt, ws);
    k_finalize<<<1, 64, 0, stream>>>(ws);
    dim3 g(BATCH / BTILE, BATCH / BTILE, 3);
    k_mmd     <<<g, 256, 0, stream>>>(src, tgt, sl, tl, ws);
    k_out     <<<1, 32, 0, stream>>>(ws, out);
}